// Loss_50362786513410
// MI455X (gfx1250) — compile-verified
//
#include <hip/hip_runtime.h>
#include <math.h>

typedef __attribute__((ext_vector_type(2))) float v2f;
typedef __attribute__((ext_vector_type(8))) float v8f;

constexpr int   Bsz   = 4096;
constexpr int   D     = 256;
constexpr int   Nrows = 2 * Bsz;          // 8192
constexpr int   NT    = Nrows / 16;       // 512 column tiles
constexpr float INV_T = 2.0f;             // 1 / temperature(0.5)

constexpr int WAVES          = 4;
constexpr int THREADS        = 32 * WAVES;   // 128
constexpr int ROWS_PER_BLOCK = 16 * WAVES;   // 64
constexpr int BSTRIDE        = D + 4;        // 260 floats: conflict-free b64 LDS reads

__device__ __forceinline__ const float* row_ptr(const float* o1, const float* o2, int j) {
    return (j < Bsz) ? (o1 + (size_t)j * D) : (o2 + (size_t)(j - Bsz) * D);
}

__device__ __forceinline__ void wait_async0() {
#if __has_builtin(__builtin_amdgcn_s_wait_asynccnt)
    __builtin_amdgcn_s_wait_asynccnt(0);
#else
    asm volatile("s_wait_asynccnt 0" ::: "memory");
#endif
}

// Async-copy 128 contiguous bytes (32 floats) from global to LDS.
// LDS VDST operand = low 32 bits of the generic pointer (== workgroup LDS offset,
// since local->flat addrspacecast is {aperture_hi32, lds_offset_lo32}).
__device__ __forceinline__ void async_copy_128B(const float* __restrict__ src, float* dst) {
    uint32_t ldsOff = (uint32_t)(uintptr_t)(void*)dst;
    uint64_t gaddr  = (uint64_t)(uintptr_t)src;
    asm volatile(
        "global_load_async_to_lds_b128 %0, %1, off\n\t"
        "global_load_async_to_lds_b128 %0, %1, off offset:16\n\t"
        "global_load_async_to_lds_b128 %0, %1, off offset:32\n\t"
        "global_load_async_to_lds_b128 %0, %1, off offset:48\n\t"
        "global_load_async_to_lds_b128 %0, %1, off offset:64\n\t"
        "global_load_async_to_lds_b128 %0, %1, off offset:80\n\t"
        "global_load_async_to_lds_b128 %0, %1, off offset:96\n\t"
        "global_load_async_to_lds_b128 %0, %1, off offset:112"
        :: "v"(ldsOff), "v"(gaddr) : "memory");
}

__global__ __launch_bounds__(THREADS)
void ntxent_rows_kernel(const float* __restrict__ out1,
                        const float* __restrict__ out2,
                        float* __restrict__ row_loss) {
    __shared__ float ldsB[2][16 * BSTRIDE];  // 33.3 KB: double-buffered column tiles
    __shared__ float red[WAVES][32][8];      // 4 KB: cross-lane row-sum reduction

    const int tid  = threadIdx.x;
    const int wid  = tid >> 5;
    const int lane = tid & 31;
    const int half = lane >> 4;              // 0: lanes 0-15 (K=0,1), 1: lanes 16-31 (K=2,3)
    const int l16  = lane & 15;

    const int blockRow0 = blockIdx.x * ROWS_PER_BLOCK;
    const int R0        = blockRow0 + wid * 16;   // this wave's 16-row tile

    // Per-lane A-row pointer (row = R0 + l16); lanes 16-31 read the K+2 slice.
    const float* aPtr = row_ptr(out1, out2, R0 + l16) + half * 2;

    // Each thread owns 32 floats of the 16x256 tile: row = tid>>3, cols (tid&7)*32..+31.
    const int ldRow = tid >> 3;
    const int ldCol = (tid & 7) * 32;

    float acc[8];
#pragma unroll
    for (int v = 0; v < 8; ++v) acc[v] = 0.0f;

    // Prologue: stage tile 0 into buffer 0.
    async_copy_128B(row_ptr(out1, out2, ldRow) + ldCol, &ldsB[0][ldRow * BSTRIDE + ldCol]);
    wait_async0();
    __syncthreads();

    for (int t = 0; t < NT; ++t) {
        const int C0  = t << 4;
        const int cur = t & 1;

        // Prefetch tile t+1 into the other buffer (overlaps with WMMA below).
        if (t + 1 < NT) {
            const float* src = row_ptr(out1, out2, ((t + 1) << 4) + ldRow) + ldCol;
            async_copy_128B(src, &ldsB[cur ^ 1][ldRow * BSTRIDE + ldCol]);
        }

        // 16x16 tile of logits: D = A(16xK) x B(Kx16), K swept 4 at a time.
        v8f c = {};
        const float* bPtr = &ldsB[cur][l16 * BSTRIDE + half * 2];
#pragma unroll 8
        for (int k = 0; k < D; k += 4) {
            v2f a = *(const v2f*)(aPtr + k);
            v2f b = *(const v2f*)(bPtr + k);
            c = __builtin_amdgcn_wmma_f32_16x16x4_f32(
                    /*neg_a=*/false, a, /*neg_b=*/false, b,
                    /*c_mod=*/(short)0, c, /*reuse_a=*/false, /*reuse_b=*/false);
        }

        // exp(logit / T), excluding the diagonal (only present when C0 == R0).
        const bool diagTile = (C0 == R0);
        const int  mbase    = half * 8;       // C/D layout: lane<16 -> M=v, lane>=16 -> M=v+8
#pragma unroll
        for (int v = 0; v < 8; ++v) {
            float e = __expf(INV_T * c[v]);
            if (diagTile && (v + mbase) == l16) e = 0.0f;
            acc[v] += e;
        }

        // Prefetched tile must be fully in LDS and visible to all waves.
        wait_async0();
        __syncthreads();
    }

    // Row m sum = sum over 16 lanes of one half-wave of acc[m & 7].
#pragma unroll
    for (int v = 0; v < 8; ++v) red[wid][lane][v] = acc[v];
    __syncthreads();

    if (tid < ROWS_PER_BLOCK) {
        const int w     = tid >> 4;
        const int m     = tid & 15;
        const int lbase = (m >= 8) ? 16 : 0;
        const int vv    = m & 7;
        float s = 0.0f;
#pragma unroll
        for (int l = 0; l < 16; ++l) s += red[w][lbase + l][vv];

        // loss_i = log(row_sum_i) - log(pos_i),  log(pos_i) = 2 * <out1_p, out2_p>
        const int i = blockRow0 + tid;
        const int p = (i < Bsz) ? i : i - Bsz;
        const float* r1 = out1 + (size_t)p * D;
        const float* r2 = out2 + (size_t)p * D;
        float pd = 0.0f;
        for (int k = 0; k < D; k += 4) {
            float4 x = *(const float4*)(r1 + k);
            float4 y = *(const float4*)(r2 + k);
            pd += x.x * y.x + x.y * y.y + x.z * y.z + x.w * y.w;
        }
        row_loss[i] = __logf(s) - INV_T * pd;
    }
}

__global__ __launch_bounds__(256)
void reduce_mean_kernel(const float* __restrict__ row_loss, float* __restrict__ out) {
    __shared__ float s[256];
    float p = 0.0f;
    for (int i = threadIdx.x; i < Nrows; i += 256) p += row_loss[i];
    s[threadIdx.x] = p;
    __syncthreads();
    for (int off = 128; off > 0; off >>= 1) {
        if ((int)threadIdx.x < off) s[threadIdx.x] += s[threadIdx.x + off];
        __syncthreads();
    }
    if (threadIdx.x == 0) out[0] = s[0] / (float)Nrows;
}

extern "C" void kernel_launch(void* const* d_in, const int* in_sizes, int n_in,
                              void* d_out, int out_size, void* d_ws, size_t ws_size,
                              hipStream_t stream) {
    const float* o1  = (const float*)d_in[0];
    const float* o2  = (const float*)d_in[1];
    float*       out = (float*)d_out;
    float*       ws  = (float*)d_ws;   // 8192 floats of per-row loss

    ntxent_rows_kernel<<<Nrows / ROWS_PER_BLOCK, THREADS, 0, stream>>>(o1, o2, ws);
    reduce_mean_kernel<<<1, 256, 0, stream>>>(ws, out);
}